// Loss_3238405341554
// MI455X (gfx1250) — compile-verified
//
#include <hip/hip_runtime.h>
#include <stdint.h>

// ---------------------------------------------------------------------------
// MI455X (gfx1250) trajectory-loss reduction.
//
// Bandwidth-bound: 227 MB in -> 13 floats out (~10 us at 23.3 TB/s).
// Data movement uses the CDNA5 Tensor Data Mover (tensor_load_to_lds) with a
// 2D strided descriptor (row = 60 f32, stride = 360 f32) + LDS padding
// (1 dword per 16B => LDS row stride 75 dwords, odd => bank-conflict free),
// double-buffered across the 6-mode loop and synchronized with TENSORcnt.
// ---------------------------------------------------------------------------

#define T_STEPS 30
#define NMODS   6
#define BLK     64        // threads per block == samples per block (2 waves)
#define ROWDW   60        // dwords of payload per sample row
#define ROWP    75        // padded LDS row stride in dwords (60 + 15 pad)

typedef uint32_t u32;
typedef uint32_t u32x4 __attribute__((ext_vector_type(4)));
typedef uint32_t u32x8 __attribute__((ext_vector_type(8)));

// dword index i within a row -> padded LDS dword offset (pad 1 dw per 4 dw)
__device__ __forceinline__ int pidx(int i) { return i + (i >> 2); }

template <int N>
__device__ __forceinline__ void tdm_wait() {
#if __has_builtin(__builtin_amdgcn_s_wait_tensorcnt)
  __builtin_amdgcn_s_wait_tensorcnt(N);
#else
  asm volatile("s_wait_tensorcnt %0" :: "i"(N) : "memory");
#endif
}

// 2D TDM load: tile 60 x 64 f32 elements, row stride `stride_elems`,
// LDS destination padded 1 dword per 16 bytes (pad_interval=1, pad_amount=0).
__device__ __forceinline__ void tdm_load_2d(u32 lds_byte_addr, const void* gptr,
                                            u32 rows_avail, u32 stride_elems) {
  uint64_t ga = (uint64_t)(uintptr_t)gptr;
  u32x4 g0;
  g0[0] = 0x1u;                                           // count=1, user mode
  g0[1] = lds_byte_addr;                                  // lds_addr
  g0[2] = (u32)ga;                                        // global_addr[31:0]
  g0[3] = ((u32)(ga >> 32) & 0x01FFFFFFu) | (2u << 30);   // ga[56:32] | type=2
  u32x8 g1;
  g1[0] = (2u << 16)        // data_size = 4B
        | (1u << 20)        // pad_enable
        | (1u << 22);       // pad_interval = 16B (pad_amount=0 -> 1 dword)
  g1[1] = ((u32)ROWDW) << 16;                 // tensor_dim0[15:0] = 60
  g1[2] = (rows_avail & 0xFFFFu) << 16;       // dim0 hi=0 | tensor_dim1 lo16
  g1[3] = (rows_avail >> 16) | (((u32)ROWDW) << 16); // dim1 hi | tile_dim0=60
  g1[4] = (u32)BLK;                           // tile_dim1 = 64, tile_dim2 = 0
  g1[5] = stride_elems;                       // tensor_dim0_stride[31:0]
  g1[6] = 0u;                                 // stride hi | dim1_stride lo (2D)
  g1[7] = 0u;
  asm volatile("tensor_load_to_lds %0, %1" :: "s"(g0), "s"(g1) : "memory");
}

__global__ __launch_bounds__(BLK) void traj_loss_kernel(
    const float* __restrict__ cls, const float* __restrict__ reg,
    const float* __restrict__ gt,  const unsigned char* __restrict__ has,
    float* __restrict__ out, int B) {
  __shared__ __align__(16) float s_gt[BLK * ROWP];
  __shared__ __align__(16) float s_reg[2][BLK * ROWP];
  __shared__ float s_cls[BLK * NMODS];
  __shared__ float s_part[2 * 12];

  const int tid = threadIdx.x;
  const int b0  = blockIdx.x * BLK;
  const int b   = b0 + tid;
  const bool live = (b < B);
  const float gate = live ? 1.0f : 0.0f;

  // --- stage cls cooperatively (coalesced) ---------------------------------
  for (int i = tid; i < BLK * NMODS; i += BLK) {
    int gi = b0 * NMODS + i;
    s_cls[i] = (gi < B * NMODS) ? cls[gi] : 0.0f;
  }

  // --- wave 0 issues TDM loads (scalar branch: TDM ignores EXEC) -----------
  if (__builtin_amdgcn_readfirstlane(tid) == 0) {
    u32 rows = (u32)(B - b0);
    tdm_load_2d((u32)(uintptr_t)&s_gt[0],     gt  + (size_t)b0 * 60,       rows, 60u);
    tdm_load_2d((u32)(uintptr_t)&s_reg[0][0], reg + (size_t)b0 * 360,      rows, 360u);
    tdm_load_2d((u32)(uintptr_t)&s_reg[1][0], reg + (size_t)b0 * 360 + 60, rows, 360u);
    tdm_wait<1>();  // gt + mode0 complete (in-order); mode1 may still fly
  }

  // --- per-lane: has mask + last index (no LDS dependency) -----------------
  u32 hmask = 0;
  if (live) {
    const unsigned char* hp = has + (size_t)b * T_STEPS;
#pragma unroll
    for (int t = 0; t < T_STEPS; ++t) hmask |= ((u32)(hp[t] != 0)) << t;
  }
  int last_idx = 0;
  float last_max = -1e30f;
#pragma unroll
  for (int t = 0; t < T_STEPS; ++t) {
    float hv = (float)((hmask >> t) & 1u);
    float v = hv + 0.1f * ((float)t / 30.0f);
    if (v > last_max) { last_max = v; last_idx = t; }  // first max (strict >)
  }
  const float valid = (last_max > 1.0f) ? 1.0f : 0.0f;

  __syncthreads();  // publish s_gt / s_reg[0] / s_cls

  const float* gr = &s_gt[tid * ROWP];

  // --- heading -> rotation (cos/sin) kept in VGPRs -------------------------
  float ct[T_STEPS], st[T_STEPS];
  {
    float ang[T_STEPS - 1];
#pragma unroll
    for (int t = 0; t < T_STEPS - 1; ++t) {
      float x0 = gr[pidx(2 * t)],     y0 = gr[pidx(2 * t + 1)];
      float x1 = gr[pidx(2 * t + 2)], y1 = gr[pidx(2 * t + 3)];
      ang[t] = atan2f(y1 - y0, x1 - x0);
    }
    float dx = gr[pidx(0)] - gr[pidx(2 * (T_STEPS - 1))];
    float dy = gr[pidx(1)] - gr[pidx(2 * (T_STEPS - 1) + 1)];
    bool cond = sqrtf(dx * dx + dy * dy) > 2.0f;
    const float R2D = 57.29577951308232f;
    const float D2R = 0.017453292519943295f;
    float head[T_STEPS];
    head[0]           = ang[0] * R2D;
    head[T_STEPS - 1] = ang[T_STEPS - 2] * R2D;
#pragma unroll
    for (int t = 1; t < T_STEPS - 1; ++t) {
      float f = ang[t], w = ang[t - 1];
      float tmp = f * R2D + w * R2D;
      head[t] = (f == 0.0f || w == 0.0f) ? tmp : tmp * 0.5f;
    }
#pragma unroll
    for (int t = 0; t < T_STEPS; ++t) {
      float th = cond ? (-head[t] * D2R) : 0.0f;
      ct[t] = cosf(th);
      st[t] = sinf(th);
    }
  }
  const float glx = gr[pidx(2 * last_idx)];
  const float gly = gr[pidx(2 * last_idx + 1)];

  // --- 6-mode pass over LDS-staged reg, double buffered via TDM ------------
  float distm[NMODS], sxm[NMODS], sym[NMODS], fxm[NMODS], fym[NMODS], sl1m[NMODS];
#pragma unroll
  for (int m = 0; m < NMODS; ++m) {
    const float* rr = &s_reg[m & 1][tid * ROWP];
    float sx = 0.f, sy = 0.f, sl1 = 0.f, fx = 0.f, fy = 0.f, rlx = 0.f, rly = 0.f;
#pragma unroll
    for (int t = 0; t < T_STEPS; ++t) {
      float rx = rr[pidx(2 * t)], ry = rr[pidx(2 * t + 1)];
      float gx = gr[pidx(2 * t)], gy = gr[pidx(2 * t + 1)];
      float ex = fabsf(gx - rx), ey = fabsf(gy - ry);
      float dex = fabsf(ct[t] * ex - st[t] * ey);
      float dey = fabsf(st[t] * ex + ct[t] * ey);
      sx += dex; sy += dey;
      if (t == T_STEPS - 1) { fx = dex; fy = dey; }
      float dxx = rx - gx, dyy = ry - gy;
      float ax = fabsf(dxx), ay = fabsf(dyy);
      float slx = (ax < 1.0f) ? 0.5f * dxx * dxx : ax - 0.5f;
      float sly = (ay < 1.0f) ? 0.5f * dyy * dyy : ay - 0.5f;
      sl1 += ((float)((hmask >> t) & 1u)) * (slx + sly);
      if (t == last_idx) { rlx = rx; rly = ry; }
    }
    float ddx = rlx - glx, ddy = rly - gly;
    distm[m] = sqrtf(ddx * ddx + ddy * ddy);
    sxm[m] = sx; sym[m] = sy; fxm[m] = fx; fym[m] = fy; sl1m[m] = sl1;

    __syncthreads();  // all waves done reading buffer (m&1)
    if (__builtin_amdgcn_readfirstlane(tid) == 0) {
      if (m + 2 < NMODS) {
        tdm_load_2d((u32)(uintptr_t)&s_reg[m & 1][0],
                    reg + (size_t)b0 * 360 + (size_t)(m + 2) * 60,
                    (u32)(B - b0), 360u);
        tdm_wait<1>();  // mode m+1 done (in-order), m+2 may still fly
      } else {
        tdm_wait<0>();
      }
    }
    __syncthreads();  // publish buffer ((m+1)&1)
  }

  // --- per-sample finalization ---------------------------------------------
  float min_dist = distm[0]; int min_m = 0;
#pragma unroll
  for (int m = 1; m < NMODS; ++m)
    if (distm[m] < min_dist) { min_dist = distm[m]; min_m = m; }  // first min

  float clsv[NMODS];
#pragma unroll
  for (int m = 0; m < NMODS; ++m) clsv[m] = s_cls[tid * NMODS + m];

  float cls_min = clsv[0], sl1_best = sl1m[0];
#pragma unroll
  for (int m = 1; m < NMODS; ++m)
    if (m == min_m) { cls_min = clsv[m]; sl1_best = sl1m[m]; }

  int top1 = 0; float cmax = clsv[0];
#pragma unroll
  for (int m = 1; m < NMODS; ++m)
    if (clsv[m] > cmax) { cmax = clsv[m]; top1 = m; }  // first max

  float wcnt = 0.f, mgnw = 0.f;
#pragma unroll
  for (int m = 0; m < NMODS; ++m) {
    float mgn = cls_min - clsv[m];
    bool w = (min_dist < 2.0f) && ((distm[m] - min_dist) > 0.2f) &&
             (valid > 0.0f) && (mgn < 0.2f);
    if (w) { wcnt += 1.0f; mgnw += mgn; }
  }

  float a6x = sxm[0], a6y = sym[0], f6x = fxm[0], f6y = fym[0];
#pragma unroll
  for (int m = 1; m < NMODS; ++m) {
    a6x = fminf(a6x, sxm[m]); a6y = fminf(a6y, sym[m]);
    f6x = fminf(f6x, fxm[m]); f6y = fminf(f6y, fym[m]);
  }
  float a1x = sxm[0], a1y = sym[0], f1x = fxm[0], f1y = fym[0];
#pragma unroll
  for (int m = 1; m < NMODS; ++m)
    if (m == top1) { a1x = sxm[m]; a1y = sym[m]; f1x = fxm[m]; f1y = fym[m]; }

  float acc[12];
  acc[0]  = gate * (0.2f * wcnt - mgnw);          // cls_loss contribution
  acc[1]  = gate * wcnt;                          // num_cls
  acc[2]  = gate * valid * sl1_best;              // reg_loss contribution
  acc[3]  = gate * valid * (float)__popc(hmask);  // num_reg
  acc[4]  = gate * a6x;  acc[5]  = gate * a6y;
  acc[6]  = gate * f6x;  acc[7]  = gate * f6y;
  acc[8]  = gate * a1x;  acc[9]  = gate * a1y;
  acc[10] = gate * f1x;  acc[11] = gate * f1y;

  // --- wave32 tree reduce, then one atomic set per block -------------------
#pragma unroll
  for (int k = 0; k < 12; ++k) {
    float v = acc[k];
#pragma unroll
    for (int off = 16; off > 0; off >>= 1) v += __shfl_xor(v, off, 32);
    if ((tid & 31) == 0) s_part[(tid >> 5) * 12 + k] = v;
  }
  __syncthreads();
  if (tid < 12) atomicAdd(&out[tid + 1], s_part[tid] + s_part[12 + tid]);
}

__global__ void init_out_kernel(float* out) {
  if (threadIdx.x < 13) out[threadIdx.x] = 0.0f;
}

__global__ void finalize_kernel(float* out) {
  if (threadIdx.x == 0)
    out[0] = out[1] / (out[2] + 1e-10f) + out[3] / (out[4] + 1e-10f);
}

extern "C" void kernel_launch(void* const* d_in, const int* in_sizes, int n_in,
                              void* d_out, int out_size, void* d_ws, size_t ws_size,
                              hipStream_t stream) {
  const float* cls          = (const float*)d_in[0];
  const float* reg          = (const float*)d_in[1];
  const float* gt           = (const float*)d_in[2];
  const unsigned char* has  = (const unsigned char*)d_in[3];
  float* out = (float*)d_out;

  int B = in_sizes[0] / NMODS;
  int grid = (B + BLK - 1) / BLK;

  init_out_kernel<<<1, 32, 0, stream>>>(out);
  traj_loss_kernel<<<grid, BLK, 0, stream>>>(cls, reg, gt, has, out, B);
  finalize_kernel<<<1, 32, 0, stream>>>(out);
}